// CTCCompressor_72078141162116
// MI455X (gfx1250) — compile-verified
//
#include <hip/hip_runtime.h>
#include <hip/hip_bf16.h>

// Problem constants from the reference: B=16, T=2048, D=512, V=4096.
#define B_ 16
#define T_ 2048
#define D_ 512
#define V_ 4096
#define NGROUP (T_ / 16)   // 128 groups of 16 segments per batch
#define NDCH   (D_ / 16)   // 32 chunks of 16 dims
#define SWPAD  4           // sentinel tail so K3 never needs a range check
#define SWSTR  (T_ + SWPAD)

typedef __attribute__((ext_vector_type(2))) float v2f;
typedef __attribute__((ext_vector_type(4))) float v4f;
typedef __attribute__((ext_vector_type(8))) float v8f;

union Acc8 { v8f v; float f[8]; };

// ---------------------------------------------------------------------------
// K1: one wave per (b,t) row. Online softmax over V=4096:
//   m = max, s = sum exp(x-m), amax = argmax (first occurrence on ties).
//   p = 1/s  (probability of the argmax token).
// Streams 512 MiB exactly once -> non-temporal b128 loads (don't pollute L2;
// keep the 192MB L2 for hidden/sw which K3 re-reads across 32 dim-chunk waves).
// Roofline-dominant kernel: ~22us at 23.3 TB/s.
// ---------------------------------------------------------------------------
__global__ __launch_bounds__(256) void ctc_rowstat_kernel(
    const float* __restrict__ logits, int* __restrict__ pred, float* __restrict__ pOut)
{
    const int lane = threadIdx.x & 31;
    const int wave = threadIdx.x >> 5;
    const int row  = blockIdx.x * 8 + wave;          // flat b*T + t
    const float* __restrict__ rp = logits + (size_t)row * V_;

    float m = -3.4028235e38f;
    float s = 0.f;
    int   amax = 0;

    for (int it = 0; it < V_ / 128; ++it) {
        const int base = it * 128 + lane * 4;        // 512B per wave-iteration
        const v4f q = __builtin_nontemporal_load((const v4f*)(rp + base));
        #pragma unroll
        for (int j = 0; j < 4; ++j) {
            const float x = q[j];
            if (x > m) { s = s * __expf(m - x) + 1.f; m = x; amax = base + j; }
            else       { s += __expf(x - m); }
        }
    }
    // wave32 butterfly combine of (m, s, amax); min-index tie-break == jnp.argmax
    #pragma unroll
    for (int off = 1; off < 32; off <<= 1) {
        const float om = __shfl_xor(m, off, 32);
        const float os = __shfl_xor(s, off, 32);
        const int   oi = __shfl_xor(amax, off, 32);
        const float M  = fmaxf(m, om);
        const float ns = s * __expf(m - M) + os * __expf(om - M);
        int ni;
        if (om > m)      ni = oi;
        else if (om < m) ni = amax;
        else             ni = (oi < amax) ? oi : amax;
        m = M; s = ns; amax = ni;
    }
    if (lane == 0) { pred[row] = amax; pOut[row] = 1.f / s; }
}

// ---------------------------------------------------------------------------
// K2: one 256-thread block per batch.
//  - start flags, block-wide inclusive scan (LDS) -> segment ids
//  - segFirst[s] = first frame of segment s; segFirst[new_len] = len
//  - per-segment sum of p via LDS ds_add_f32 atomics (2048 floats = 8KB of
//    the 320KB/WGP LDS)
//  - per-frame {seg, w} packed into one int2 (single b64 load in K3);
//    invalid frames and the 4 tail slots get the sentinel {-1, 0} so K3's
//    hit test needs NO range check (and thus cannot be branch-ified).
//  - writes new_lengths into d_out tail (int bits)
// ---------------------------------------------------------------------------
__global__ __launch_bounds__(256) void ctc_segment_kernel(
    const int* __restrict__ pred, const float* __restrict__ p,
    const int* __restrict__ lengths,
    int2* __restrict__ sw,
    int* __restrict__ segFirst, int* __restrict__ newLen,
    int* __restrict__ outLenTail)
{
    __shared__ float segp[T_];
    __shared__ int   sh[256];

    const int b   = blockIdx.x;
    const int i   = threadIdx.x;
    const int len = lengths[b];

    for (int k = i; k < T_; k += 256) segp[k] = 0.f;   // zeroed before any sync

    // 8 consecutive frames per thread: start flags
    const int t0 = i * 8;
    int prev = (t0 > 0) ? pred[b * T_ + t0 - 1] : -1;
    int cnt = 0; unsigned flags = 0u;
    #pragma unroll
    for (int j = 0; j < 8; ++j) {
        const int t  = t0 + j;
        const int pr = pred[b * T_ + t];
        const bool st = (t < len) && (t == 0 || pr != prev);
        prev = pr;
        if (st) { flags |= 1u << j; ++cnt; }
    }
    sh[i] = cnt;
    __syncthreads();
    for (int off = 1; off < 256; off <<= 1) {          // Hillis-Steele scan
        const int v = (i >= off) ? sh[i - off] : 0;
        __syncthreads();
        sh[i] += v;
        __syncthreads();
    }
    const int total = sh[255];                         // new_lengths[b]
    int run = sh[i] - cnt;                             // exclusive prefix
    int segl[8];
    #pragma unroll
    for (int j = 0; j < 8; ++j) {
        const int t = t0 + j;
        if (flags & (1u << j)) { segFirst[b * (T_ + 1) + run] = t; ++run; }
        segl[j] = run - 1;
    }
    if (i == 0) {
        newLen[b] = total;
        segFirst[b * (T_ + 1) + total] = len;          // sentinel end-of-last-group
        outLenTail[b] = total;                         // tuple output #2
        #pragma unroll
        for (int j = 0; j < SWPAD; ++j) {              // tail sentinels
            int2 q; q.x = -1; q.y = 0;
            sw[b * SWSTR + T_ + j] = q;
        }
    }
    // seg_p via LDS float atomics (ds_add_f32)
    #pragma unroll
    for (int j = 0; j < 8; ++j) {
        const int t = t0 + j;
        if (t < len) atomicAdd(&segp[segl[j]], p[b * T_ + t]);
    }
    __syncthreads();
    #pragma unroll
    for (int j = 0; j < 8; ++j) {
        const int t = t0 + j;
        int2 q;
        if (t < len) {
            q.x = segl[j];
            q.y = __float_as_int(p[b * T_ + t] / (segp[segl[j]] + 1e-10f));
        } else {
            q.x = -1;                                  // sentinel: never matches
            q.y = 0;
        }
        sw[b * SWSTR + t] = q;                         // one b64 store
    }
}

// ---------------------------------------------------------------------------
// K3: segmented weighted sum as WMMA f32 16x16x4 GEMM.
// One wave per (batch, 16-segment group, 16-dim chunk).
//   Out(16 segs x 16 dims) += A(16 segs x 4 frames) * B(4 frames x 16 dims)
// A built in-register: A[m][k] = (seg[tk]==segbase+m) ? w[tk] : 0.
// Frame ranges of the groups tile [0,len) disjointly -> hidden read exactly once.
// Truly branchless inner loop:
//  - SW[tk] is ALWAYS in-bounds (sentinel-padded stride T+4), so there is no
//    range predicate at all; the hit test depends only on the loaded seg id,
//    which the compiler cannot use to predicate its own load.
//  - frames of the next group have seg >= segbase+16, invalid frames seg=-1:
//    both auto-miss, so out-of-range K columns of A are zero and the unmasked
//    B row (hidden clamped to row T-1 with a constant v_min) contributes 0.
// Register layouts per CDNA5 ISA 7.12.2 (wave32):
//   A: lanes 0-15 hold M=0..15 with K={0,1} in v[0],v[1]; lanes 16-31: K={2,3}
//   B: VGPR j: lanes 0-15 = row K=2*hi+j cols 0-15 (mirrors A striping)
//   C/D: VGPR r: lanes 0-15 -> M=r, lanes 16-31 -> M=r+8, N=lane&15
// ---------------------------------------------------------------------------
__global__ __launch_bounds__(32) void ctc_pool_kernel(
    const float* __restrict__ hidden,
    const int2* __restrict__ sw,
    const int* __restrict__ segFirst, const int* __restrict__ newLen,
    const int* __restrict__ lengths, float* __restrict__ out)
{
    const int id  = blockIdx.x;
    const int dc  = id & (NDCH - 1);
    const int g   = (id / NDCH) & (NGROUP - 1);
    const int b   = id / (NDCH * NGROUP);
    const int lane = threadIdx.x;
    const int L   = lane & 15;
    const int hi  = lane >> 4;
    const int segbase = g * 16;
    const int d0  = dc * 16;
    const int nl  = newLen[b];
    const int len = lengths[b];

    const float* __restrict__ H  = hidden + (size_t)b * T_ * D_;
    const unsigned long long* __restrict__ SW =
        (const unsigned long long*)(sw + b * SWSTR);
    float* __restrict__ O = out + (size_t)b * T_ * D_;

    int t0 = 0, t1 = 0;
    if (segbase < nl) {
        t0 = segFirst[b * (T_ + 1) + segbase];
        const int se = segbase + 16;
        t1 = (se < nl) ? segFirst[b * (T_ + 1) + se] : len;
    }

    Acc8 c = {};   // zero accumulator

#if defined(__gfx1250__) && __has_builtin(__builtin_amdgcn_wmma_f32_16x16x4_f32)
    #pragma unroll 2
    for (int t = t0; t < t1; t += 4) {
        v2f a, bv;
        #pragma unroll
        for (int j = 0; j < 2; ++j) {
            const int tk = t + 2 * hi + j;           // K index = 2*hi + j
            // tk <= t1+2 <= len+2 <= T+2 < T+SWPAD : always in-bounds.
            const unsigned long long q = SW[tk];     // ONE global_load_b64
            const int sgv = (int)(unsigned)q;        // .x = seg id (or -1)
            const int wb  = (int)(q >> 32);          // .y = w bits
            // Hit test uses only the loaded value -> load cannot be predicated.
            a[j] = (sgv == segbase + L) ? __int_as_float(wb) : 0.f;
            const int th = (tk < T_) ? tk : (T_ - 1);   // constant v_min clamp
            bv[j] = H[(size_t)th * D_ + d0 + L];     // unmasked (A col zero)
        }
        // (neg_a, A, neg_b, B, c_mod, C, reuse_a, reuse_b)
        c.v = __builtin_amdgcn_wmma_f32_16x16x4_f32(
            false, a, false, bv, (short)0, c.v, false, false);
    }
#else
    // VALU fallback with identical C-register mapping
    for (int t = t0; t < t1; ++t) {
        const unsigned long long q = SW[t];
        const int sg = (int)(unsigned)q - segbase;
        if (sg >= 0 && sg < 16 && (sg >> 3) == hi)
            c.f[sg & 7] += __int_as_float((int)(q >> 32)) *
                           H[(size_t)t * D_ + d0 + L];
    }
#endif

    #pragma unroll
    for (int r = 0; r < 8; ++r)                      // D layout: M = r + 8*hi, N = L
        O[(size_t)(segbase + r + 8 * hi) * D_ + d0 + L] = c.f[r];
}

// ---------------------------------------------------------------------------
extern "C" void kernel_launch(void* const* d_in, const int* in_sizes, int n_in,
                              void* d_out, int out_size, void* d_ws, size_t ws_size,
                              hipStream_t stream)
{
    const float* hidden  = (const float*)d_in[0];
    const float* logits  = (const float*)d_in[1];
    const int*   lengths = (const int*)d_in[2];
    float* out = (float*)d_out;

    // Workspace layout (~650 KB total)
    char* ws = (char*)d_ws;
    int*   pred     = (int*)ws;     ws += sizeof(int)   * B_ * T_;
    float* p        = (float*)ws;   ws += sizeof(float) * B_ * T_;
    int2*  sw       = (int2*)ws;    ws += sizeof(int2)  * B_ * SWSTR;
    int*   segFirst = (int*)ws;     ws += sizeof(int)   * B_ * (T_ + 1);
    int*   newLen   = (int*)ws;     ws += sizeof(int)   * B_;
    int*   outLenTail = (int*)d_out + (size_t)B_ * T_ * D_;  // new_lengths (int bits)

    ctc_rowstat_kernel<<<B_ * T_ / 8, 256, 0, stream>>>(logits, pred, p);
    ctc_segment_kernel<<<B_, 256, 0, stream>>>(pred, p, lengths, sw,
                                               segFirst, newLen, outLenTail);
    ctc_pool_kernel<<<B_ * NGROUP * NDCH, 32, 0, stream>>>(hidden, sw,
                                                           segFirst, newLen,
                                                           lengths, out);
}